// LogMap_5866925326755
// MI455X (gfx1250) — compile-verified
//
#include <hip/hip_runtime.h>
#include <math.h>

// Problem geometry (fixed by the reference): N=512, num=25, frame=300.
// Fused inner index q = p*frame + f is contiguous in BOTH input (stride 1)
// and output rows (row = q*4 + k), so we tile over (n, q).
#define NDIM     512
#define Q_TOTAL  7500        // 25 * 300
#define STRIDE_J 7500        // j stride in input  (num*frame)
#define STRIDE_I 22500       // i stride = 3*STRIDE_J
#define STRIDE_N 67500       // n stride = 9*STRIDE_J
#define TN 32                // n-tile (lane dim on store)
#define TQ 32                // q-tile (lane dim on load)
#define PAD 33               // 33 % 64 banks -> conflict-free both phases

__global__ __launch_bounds__(256)
void logmap_transpose_kernel(const float* __restrict__ x,
                             float* __restrict__ out)
{
    // lds[k][qq][nn]: phase-1 scatter has lane stride PAD (odd -> no bank
    // conflicts); phase-2 read is stride-1 along nn (no conflicts).
    __shared__ float lds[4][TQ][PAD];

    const int t   = threadIdx.x;
    const int tq  = t & 31;               // lane -> q  (coalesced reads)
    const int tg  = t >> 5;               // wave id -> n subgroup (0..7)
    const int n0  = blockIdx.x * TN;
    const int q0  = blockIdx.y * TQ;
    const int q   = q0 + tq;
    const bool qok = (q < Q_TOTAL);

    // ---------------- Phase 1: streaming load + logmap, scatter to LDS ----
    if (qok) {
        const float* p0 = x + (size_t)(n0 + tg) * STRIDE_N + q;
#pragma unroll
        for (int sub = 0; sub < 4; ++sub) {
            const int nn = tg + sub * 8;
            const float* p = p0 + (size_t)sub * 8 * STRIDE_N;

            // Stream ahead: pull next n-subgroup's cachelines toward L2.
            if (sub < 3) __builtin_prefetch(p + 8 * STRIDE_N, 0, 0);

            // 9 coalesced 128B-per-wave non-temporal loads (zero reuse).
            const float r00 = __builtin_nontemporal_load(p + 0 * STRIDE_J);
            const float r01 = __builtin_nontemporal_load(p + 1 * STRIDE_J);
            const float r02 = __builtin_nontemporal_load(p + 2 * STRIDE_J);
            const float r10 = __builtin_nontemporal_load(p + 3 * STRIDE_J);
            const float r11 = __builtin_nontemporal_load(p + 4 * STRIDE_J);
            const float r12 = __builtin_nontemporal_load(p + 5 * STRIDE_J);
            const float r20 = __builtin_nontemporal_load(p + 6 * STRIDE_J);
            const float r21 = __builtin_nontemporal_load(p + 7 * STRIDE_J);
            const float r22 = __builtin_nontemporal_load(p + 8 * STRIDE_J);

            const float eps = 1e-12f;
            const float tr  = r00 + r11 + r22;
            const float c   = fminf(fmaxf((tr - 1.0f) * 0.5f, -1.0f), 1.0f);
            const float interior = acosf(c);
            float angle = interior;
            if (fabsf(tr - 3.0f) <= eps) angle = 0.0f;
            else if (fabsf(tr + 1.0f) <= eps) angle = 3.14159265358979323846f;

            const float s    = sinf(angle);
            const float safe = (fabsf(s) < eps) ? 1.0f : s;
            const float factor = angle / (2.0f * safe);

            // log = factor * (r - r^T); fi = [log21, log20, log10]
            float f0 = factor * (r21 - r12);
            float f1 = factor * (r20 - r02);
            float f2 = factor * (r10 - r01);
            const float inv = 1.0f / sqrtf(f0 * f0 + f1 * f1 + f2 * f2);
            f0 *= inv; f1 *= inv; f2 *= inv;

            lds[0][tq][nn] = f0;
            lds[1][tq][nn] = f1;
            lds[2][tq][nn] = f2;
            lds[3][tq][nn] = angle;
        }
    }

    __syncthreads();

    // ---------------- Phase 2: transposed streaming store -----------------
    // Output row = (q0+qq)*4 + k = q0*4 + row, contiguous along n.
    // Each wave has a uniform row -> one 128B contiguous NT store per iter.
    const int nn2 = t & 31;               // lane -> n (coalesced writes)
    const int r0  = t >> 5;               // wave id -> row subgroup
#pragma unroll
    for (int it = 0; it < 16; ++it) {
        const int row = r0 + it * 8;      // 0..127
        const int qq  = row >> 2;
        const int k   = row & 3;
        if (q0 + qq < Q_TOTAL) {
            const float v = lds[k][qq][nn2];
            __builtin_nontemporal_store(
                v, out + (size_t)(q0 * 4 + row) * NDIM + n0 + nn2);
        }
    }
}

extern "C" void kernel_launch(void* const* d_in, const int* in_sizes, int n_in,
                              void* d_out, int out_size, void* d_ws, size_t ws_size,
                              hipStream_t stream) {
    (void)in_sizes; (void)n_in; (void)d_ws; (void)ws_size; (void)out_size;
    const float* x  = (const float*)d_in[0];  // [512,3,3,25,300] f32
    float* out      = (float*)d_out;          // [30000,512] f32

    dim3 grid(NDIM / TN, (Q_TOTAL + TQ - 1) / TQ);   // (16, 235)
    logmap_transpose_kernel<<<grid, dim3(256), 0, stream>>>(x, out);
}